// TemporalFlowCell_33114197852599
// MI455X (gfx1250) — compile-verified
//
#include <hip/hip_runtime.h>
#include <hip/hip_bf16.h>

// TemporalFlowCell forward for MI455X (gfx1250, wave32, WMMA).
//
// Algebraic restructuring: the per-step rotation diag(mag*e^{i*omega}) is
// folded into the resonance matrix (row scaling), and beta only ever enters
// the recurrence through beta_t @ R, so we precompute Mt = R^T * W_in and
// produce betaR = x @ Mt^T directly. The serial scan then becomes
//    sr' = sr@Rc - si@Rs + betaR_t ;  si' = sr@Rs + si@Rc
// i.e. pure WMMA accumulation chains with one barrier per step and
// register-pipelined betaR loads.

#define BB 16
#define TT 2048
#define DD 1024
#define KK 128

typedef __attribute__((ext_vector_type(2))) float v2f;
typedef __attribute__((ext_vector_type(8))) float v8f;

// ---------------------------------------------------------------------------
// Kernel 0: Mt[n][d] = sum_k resonance[k][n] * W_in[k][d]   (128 x 1024)
// grid 64 (d tiles), 8 waves (n tiles).
// ---------------------------------------------------------------------------
__global__ __launch_bounds__(256)
void tfc_precomp(const float* __restrict__ Win,
                 const float* __restrict__ R,
                 float* __restrict__ Mt) {
    const int lane = threadIdx.x & 31;
    const int wave = threadIdx.x >> 5;     // 0..7 -> n tile
    const int d0   = blockIdx.x * 16;
    const int n0   = wave * 16;
    const int row  = lane & 15;
    const int bk   = (lane >> 4) * 2;

    v8f acc = {};
#pragma unroll
    for (int kk = 0; kk < 32; ++kk) {
        const int kd = kk * 4 + bk;
        v2f a, b;
        a.x = R[(size_t)kd * KK + n0 + row];        // A[n][k] = R[k][n]
        a.y = R[(size_t)(kd + 1) * KK + n0 + row];
        b.x = Win[(size_t)kd * DD + d0 + row];      // B[k][d] = Win[k][d]
        b.y = Win[(size_t)(kd + 1) * DD + d0 + row];
        acc = __builtin_amdgcn_wmma_f32_16x16x4_f32(
            false, a, false, b, (short)0, acc, false, false);
    }
    const int dcol = d0 + (lane & 15);
    const int nb   = n0 + ((lane >> 4) << 3);
#pragma unroll
    for (int r = 0; r < 8; ++r)
        Mt[(size_t)(nb + r) * DD + dcol] = acc[r];
}

// ---------------------------------------------------------------------------
// Kernel 1: betaR[m][n] = sum_d x[m][d] * Mt[n][d]
// block = 256 threads (8 waves). blockIdx.x = M tile (16 rows). wave = N tile.
// ---------------------------------------------------------------------------
__global__ __launch_bounds__(256)
void tfc_gemm_beta(const float* __restrict__ x,
                   const float* __restrict__ Mt,
                   float* __restrict__ betaR) {
    const int lane  = threadIdx.x & 31;
    const int wave  = threadIdx.x >> 5;
    const int m0    = blockIdx.x * 16;
    const int n0    = wave * 16;
    const int row   = lane & 15;
    const int khalf = (lane >> 4) * 2;

    const float* __restrict__ xrow = x  + (size_t)(m0 + row) * DD;
    const float* __restrict__ wrow = Mt + (size_t)(n0 + row) * DD;

    v8f acc = {};
#pragma unroll 8
    for (int k0 = 0; k0 < DD; k0 += 4) {
        v2f a = *(const v2f*)(xrow + k0 + khalf);
        v2f b = *(const v2f*)(wrow + k0 + khalf);
        acc = __builtin_amdgcn_wmma_f32_16x16x4_f32(
            false, a, false, b, (short)0, acc, false, false);
    }
    const int nn    = n0 + (lane & 15);
    const int mbase = m0 + ((lane >> 4) << 3);
#pragma unroll
    for (int r = 0; r < 8; ++r)
        betaR[(size_t)(mbase + r) * KK + nn] = acc[r];
}

// ---------------------------------------------------------------------------
// Kernel 2: serial scan. Single block, 512 threads (16 waves).
//   waves 0-7 : sr' tile = sr@Rc - si@Rs + betaR_t
//   waves 8-15: si' tile = sr@Rs + si@Rc
// Rotation-folded B fragments live in registers; state ping-pongs in LDS with
// one barrier per step; betaR is register double-buffered across steps.
// ---------------------------------------------------------------------------
__global__ __launch_bounds__(512)
void tfc_scan(const float* __restrict__ betaR,
              const float* __restrict__ alpha,
              const float* __restrict__ omega,
              const float* __restrict__ resonance,
              float* __restrict__ states,
              float* __restrict__ state_re_out,
              float* __restrict__ state_im_out) {
    __shared__ float st[2][2][BB * KK];    // [ping-pong][re/im][B*K]

    const int tid   = threadIdx.x;
    const int lane  = tid & 31;
    const int wave  = tid >> 5;
    const bool is_re = (wave < 8);
    const int ntile = (wave & 7) * 16;
    const int bn    = ntile + (lane & 15);
    const int bk    = (lane >> 4) * 2;
    const int arow  = lane & 15;
    const int mb    = (lane >> 4) << 3;

    // zero initial state buffer (st[0][*][*] is contiguous)
    for (int i = tid; i < 2 * BB * KK; i += 512)
        st[0][0][i] = 0.0f;

    // Build rotation-folded resonance fragments.
    v2f bf0[32], bf1[32];
#pragma unroll
    for (int kk = 0; kk < 32; ++kk) {
        const int kd = kk * 4 + bk;
        const float mg0 = 1.0f / (1.0f + __expf(-alpha[kd]));
        const float mg1 = 1.0f / (1.0f + __expf(-alpha[kd + 1]));
        const float c0 = __cosf(omega[kd]),  c1 = __cosf(omega[kd + 1]);
        const float s0 = __sinf(omega[kd]),  s1 = __sinf(omega[kd + 1]);
        const float r0 = resonance[(size_t)kd * KK + bn];
        const float r1 = resonance[(size_t)(kd + 1) * KK + bn];
        v2f fc, fs;
        fc.x = mg0 * c0 * r0;  fc.y = mg1 * c1 * r1;
        fs.x = mg0 * s0 * r0;  fs.y = mg1 * s1 * r1;
        if (is_re) { bf0[kk] = fc; bf1[kk].x = -fs.x; bf1[kk].y = -fs.y; }
        else       { bf0[kk] = fs; bf1[kk] = fc; }
    }

    // betaR register pipeline (real waves only)
    float bcur[8], bnext[8];
#pragma unroll
    for (int r = 0; r < 8; ++r) { bcur[r] = 0.0f; bnext[r] = 0.0f; }
    if (is_re) {
#pragma unroll
        for (int r = 0; r < 8; ++r)
            bcur[r] = betaR[((size_t)(mb + r) * TT) * KK + bn];
    }
    __syncthreads();

    for (int t = 0; t < TT; ++t) {
        // issue next-step betaR loads; consumed after the WMMA chain
        if (is_re && (t + 1 < TT)) {
#pragma unroll
            for (int r = 0; r < 8; ++r)
                bnext[r] = betaR[((size_t)(mb + r) * TT + (t + 1)) * KK + bn];
        }

        const int p = t & 1;
        const float* a_re = st[p][0];
        const float* a_im = st[p][1];
        v8f acc0 = {}, acc1 = {};
#pragma unroll
        for (int kk = 0; kk < 32; ++kk) {
            v2f a0 = *(const v2f*)(a_re + arow * KK + kk * 4 + bk);
            v2f a1 = *(const v2f*)(a_im + arow * KK + kk * 4 + bk);
            acc0 = __builtin_amdgcn_wmma_f32_16x16x4_f32(
                false, a0, false, bf0[kk], (short)0, acc0, false, false);
            acc1 = __builtin_amdgcn_wmma_f32_16x16x4_f32(
                false, a1, false, bf1[kk], (short)0, acc1, false, false);
        }

        float* dst = st[p ^ 1][is_re ? 0 : 1];
        if (is_re) {
#pragma unroll
            for (int r = 0; r < 8; ++r) {
                const float v = acc0[r] + acc1[r] + bcur[r];
                dst[(mb + r) * KK + bn] = v;
                states[((size_t)(mb + r) * TT + t) * KK + bn] = v;
                bcur[r] = bnext[r];
            }
        } else {
#pragma unroll
            for (int r = 0; r < 8; ++r)
                dst[(mb + r) * KK + bn] = acc0[r] + acc1[r];
        }
        __syncthreads();
    }

    // final states live in st[TT & 1] (TT even -> buffer 0)
    for (int i = tid; i < BB * KK; i += 512) {
        state_re_out[i] = st[TT & 1][0][i];
        state_im_out[i] = st[TT & 1][1][i];
    }
}

// ---------------------------------------------------------------------------
// Kernel 3: h = states @ W_out^T + b_out, then LayerNorm over D.
// ---------------------------------------------------------------------------
__global__ __launch_bounds__(512)
void tfc_out_ln(const float* __restrict__ states,
                const float* __restrict__ Wout,
                const float* __restrict__ b_out,
                const float* __restrict__ ln_g,
                const float* __restrict__ ln_b,
                float* __restrict__ out) {
    __shared__ float h[16 * DD];       // 64 KB

    const int lane  = threadIdx.x & 31;
    const int wave  = threadIdx.x >> 5;
    const int m0    = blockIdx.x * 16;
    const int row   = lane & 15;
    const int khalf = (lane >> 4) * 2;
    const int mb    = (lane >> 4) << 3;

    const float* __restrict__ arow_p = states + (size_t)(m0 + row) * KK;
    v2f afrag[32];
#pragma unroll
    for (int kk = 0; kk < 32; ++kk)
        afrag[kk] = *(const v2f*)(arow_p + kk * 4 + khalf);

    for (int nt = 0; nt < 4; ++nt) {
        const int n0 = (wave * 4 + nt) * 16;
        const float* __restrict__ brow = Wout + (size_t)(n0 + row) * KK;
        v8f acc = {};
#pragma unroll
        for (int kk = 0; kk < 32; ++kk) {
            v2f b = *(const v2f*)(brow + kk * 4 + khalf);
            acc = __builtin_amdgcn_wmma_f32_16x16x4_f32(
                false, afrag[kk], false, b, (short)0, acc, false, false);
        }
        const int d    = n0 + (lane & 15);
        const float bi = b_out[d];
#pragma unroll
        for (int r = 0; r < 8; ++r)
            h[(mb + r) * DD + d] = acc[r] + bi;
    }
    __syncthreads();

    const float* hrow = h + wave * DD;
    float s = 0.0f, s2 = 0.0f;
#pragma unroll 4
    for (int j = lane; j < DD; j += 32) {
        const float v = hrow[j];
        s += v; s2 += v * v;
    }
#pragma unroll
    for (int o = 16; o > 0; o >>= 1) {
        s  += __shfl_xor(s, o, 32);
        s2 += __shfl_xor(s2, o, 32);
    }
    const float mu  = s * (1.0f / DD);
    const float var = s2 * (1.0f / DD) - mu * mu;
    const float rs  = rsqrtf(var + 1e-5f);

    float* __restrict__ orow = out + (size_t)(m0 + wave) * DD;
#pragma unroll 4
    for (int j = lane; j < DD; j += 32)
        orow[j] = ln_g[j] * (hrow[j] - mu) * rs + ln_b[j];
}

// ---------------------------------------------------------------------------
extern "C" void kernel_launch(void* const* d_in, const int* in_sizes, int n_in,
                              void* d_out, int out_size, void* d_ws, size_t ws_size,
                              hipStream_t stream) {
    const float* x         = (const float*)d_in[0];
    const float* alpha     = (const float*)d_in[1];
    const float* omega     = (const float*)d_in[2];
    const float* W_in      = (const float*)d_in[3];
    const float* resonance = (const float*)d_in[4];
    const float* W_out     = (const float*)d_in[5];
    const float* b_out     = (const float*)d_in[6];
    const float* ln_g      = (const float*)d_in[7];
    const float* ln_b      = (const float*)d_in[8];

    float* out    = (float*)d_out;
    float* Mt     = (float*)d_ws;                       // K*D f32 = 0.5 MB
    float* betaR  = Mt + (size_t)KK * DD;               // B*T*K f32 = 16 MB
    float* states = betaR + (size_t)BB * TT * KK;       // B*T*K f32 = 16 MB
    float* st_re  = out + (size_t)BB * TT * DD;
    float* st_im  = st_re + BB * KK;

    tfc_precomp<<<DD / 16, 256, 0, stream>>>(W_in, resonance, Mt);
    tfc_gemm_beta<<<(BB * TT) / 16, 256, 0, stream>>>(x, Mt, betaR);
    tfc_scan<<<1, 512, 0, stream>>>(betaR, alpha, omega, resonance,
                                    states, st_re, st_im);
    tfc_out_ln<<<(BB * TT) / 16, 512, 0, stream>>>(states, W_out, b_out,
                                                   ln_g, ln_b, out);
}